// NRMEModule_83777632076242
// MI455X (gfx1250) — compile-verified
//
#include <hip/hip_runtime.h>
#include <math.h>

typedef __attribute__((ext_vector_type(2))) float v2f;
typedef __attribute__((ext_vector_type(4))) float v4f;
typedef __attribute__((ext_vector_type(8))) float v8f;

#define D_MODEL 1024
#define NUM_BLOCKS 4
#define BATCH 8
#define SEQ 4096
#define BS (SEQ / NUM_BLOCKS)          // 1024 rows per block
#define NBLK (BATCH * NUM_BLOCKS)      // 32 pooled rows
#define ROW_CHUNKS 8
#define ROWS_PER_CHUNK (BS / ROW_CHUNKS) // 128

// ---------------- Phase 1a: partial column sums over row chunks ----------------
// grid = NBLK * ROW_CHUNKS (256) blocks, 256 threads; each thread owns 4 columns.
// x reads use default (RT) temporal hint so x stays resident in the 192MB L2
// for the phase-3 re-read.
__global__ void se_pool_partial(const float* __restrict__ x, float* __restrict__ part) {
    int wg    = blockIdx.x;
    int blk   = wg / ROW_CHUNKS;   // 0..31  (b*4 + n)
    int chunk = wg % ROW_CHUNKS;   // 0..7
    int c4    = threadIdx.x;       // 0..255 -> columns 4*c4 .. 4*c4+3
    const float* base = x + (size_t)blk * BS * D_MODEL
                          + (size_t)chunk * ROWS_PER_CHUNK * D_MODEL
                          + 4 * c4;
    v4f acc = {0.f, 0.f, 0.f, 0.f};
    #pragma unroll 4
    for (int r = 0; r < ROWS_PER_CHUNK; ++r) {
        acc += *(const v4f*)(base + (size_t)r * D_MODEL);
    }
    *(v4f*)(part + (size_t)wg * D_MODEL + 4 * c4) = acc;
}

// ---------------- Phase 1b: reduce partials -> pooled means ----------------
// grid = NBLK (32) blocks, 256 threads.
__global__ void se_pool_reduce(const float* __restrict__ part, float* __restrict__ pooled) {
    int blk = blockIdx.x;
    int c4  = threadIdx.x;
    v4f acc = {0.f, 0.f, 0.f, 0.f};
    #pragma unroll
    for (int c = 0; c < ROW_CHUNKS; ++c) {
        acc += *(const v4f*)(part + ((size_t)blk * ROW_CHUNKS + c) * D_MODEL + 4 * c4);
    }
    acc *= (1.0f / (float)BS);
    *(v4f*)(pooled + (size_t)blk * D_MODEL + 4 * c4) = acc;
}

// ---------------- Phase 2: gate = sigmoid(pooled @ W^T + b) via fp32 WMMA ----------------
// gate[m, e] = sigmoid( sum_d pooled[m,d] * W[e,d] + b[e] )
// M=32 (2 tiles of 16), N=1024 (64 tiles), K=1024 (256 steps of 4).
// One wave per 16x16 output tile -> 128 waves -> 16 blocks x 256 threads.
__global__ void se_gate_wmma(const float* __restrict__ pooled,
                             const float* __restrict__ W,
                             const float* __restrict__ bias,
                             float* __restrict__ gate) {
    int wave = (int)((blockIdx.x * blockDim.x + threadIdx.x) >> 5); // 0..127
    int lane = threadIdx.x & 31;
    int mt = wave & 1;        // 0..1
    int nt = wave >> 1;       // 0..63

    // A layout (f32 16x4): lanes 0-15 hold M=lane, K={0,1}; lanes 16-31 hold M=lane-16, K={2,3}
    int mrow  = mt * 16 + (lane & 15);
    int nrow  = nt * 16 + (lane & 15);     // W row index = output feature (B[k][n] = W[n,k])
    int khalf = (lane >> 4) * 2;           // 0 for lanes 0-15, 2 for lanes 16-31

    const float* aptr = pooled + (size_t)mrow * D_MODEL + khalf;
    const float* bptr = W      + (size_t)nrow * D_MODEL + khalf;

    v8f acc = {};
    #pragma unroll 4
    for (int k = 0; k < D_MODEL; k += 4) {
        v2f a = *(const v2f*)(aptr + k);
        v2f b = *(const v2f*)(bptr + k);
        acc = __builtin_amdgcn_wmma_f32_16x16x4_f32(
            /*neg_a=*/false, a, /*neg_b=*/false, b,
            /*c_mod=*/(short)0, acc, /*reuse_a=*/false, /*reuse_b=*/false);
    }

    // D layout: VGPR r -> M = mt*16 + r (lanes 0-15) or mt*16 + 8 + r (lanes 16-31), N = nt*16 + (lane&15)
    int m0 = mt * 16 + ((lane >> 4) << 3);
    int n  = nt * 16 + (lane & 15);
    float bb = bias[n];
    #pragma unroll
    for (int r = 0; r < 8; ++r) {
        float v = acc[r] + bb;
        // sigmoid via hardware rcp (1 ulp) instead of the IEEE div sequence
        float e = __expf(-v);
        gate[(size_t)(m0 + r) * D_MODEL + n] = __builtin_amdgcn_rcpf(1.0f + e);
    }
}

// ---------------- Phase 3: out = x * gate (broadcast over block length) ----------------
// Flat float4 streaming kernel. row = f>>10 (D=1024), block = row>>10 (BS=1024).
// x load hits L2 (resident from phase 1); out store is non-temporal so the
// 128 MiB of output does not evict x from L2 mid-pass.
__global__ void se_apply_gate(const float* __restrict__ x,
                              const float* __restrict__ gate,
                              float* __restrict__ out) {
    size_t i = (size_t)blockIdx.x * blockDim.x + threadIdx.x; // float4 index
    size_t f = i * 4;
    int    d   = (int)(f & (D_MODEL - 1));
    size_t row = f >> 10;
    size_t blk = row >> 10;
    v4f xv = *(const v4f*)(x + f);
    v4f gv = *(const v4f*)(gate + blk * D_MODEL + d);
    __builtin_nontemporal_store(xv * gv, (v4f*)(out + f));
}

extern "C" void kernel_launch(void* const* d_in, const int* in_sizes, int n_in,
                              void* d_out, int out_size, void* d_ws, size_t ws_size,
                              hipStream_t stream) {
    const float* x = (const float*)d_in[0];
    const float* W = (const float*)d_in[1];
    const float* b = (const float*)d_in[2];
    // d_in[3] = num_blocks scalar; fixed at compile time (4).
    float* out = (float*)d_out;

    float* ws     = (float*)d_ws;
    float* pooled = ws;                                  // NBLK * D_MODEL
    float* gate   = pooled + (size_t)NBLK * D_MODEL;     // NBLK * D_MODEL
    float* part   = gate   + (size_t)NBLK * D_MODEL;     // NBLK * ROW_CHUNKS * D_MODEL

    se_pool_partial<<<NBLK * ROW_CHUNKS, 256, 0, stream>>>(x, part);
    se_pool_reduce <<<NBLK,              256, 0, stream>>>(part, pooled);
    se_gate_wmma   <<<(NBLK / 16) * (D_MODEL / 16) / 8, 256, 0, stream>>>(pooled, W, b, gate);

    const size_t total4 = (size_t)BATCH * SEQ * D_MODEL / 4;  // 8,388,608
    se_apply_gate<<<(unsigned)(total4 / 256), 256, 0, stream>>>(x, gate, out);
}